// GNNLayer_39934605918864
// MI455X (gfx1250) — compile-verified
//
#include <hip/hip_runtime.h>
#include <hip/hip_bf16.h>

typedef __attribute__((ext_vector_type(2))) float v2f;
typedef __attribute__((ext_vector_type(8))) float v8f;

#define B_   16
#define T_   12
#define N_   300
#define D_   64
#define T2_  10
#define PAD  68           // LDS row stride in floats (conflict-free for (4*row + k) pattern)
#define NTILE 19          // ceil(300/16)

// D = A(16x4) * B(4x16) + C(16x16), full fp32 WMMA (CDNA5)
__device__ __forceinline__ v8f wmma4(v2f a, v2f b, v8f c) {
    return __builtin_amdgcn_wmma_f32_16x16x4_f32(
        /*neg_a=*/false, a, /*neg_b=*/false, b,
        /*c_mod=*/(short)0, c, /*reuse_a=*/false, /*reuse_b=*/false);
}

// Kernel 0: transpose the 64x64 FFN weights so WMMA B-operands are contiguous v2f loads.
__global__ __launch_bounds__(256) void wT_kernel(const float* __restrict__ w1,
                                                 const float* __restrict__ w2,
                                                 float* __restrict__ w1T,
                                                 float* __restrict__ w2T) {
    const int i = blockIdx.x * 256 + threadIdx.x;   // 4096 total
    const int r = i >> 6, c = i & 63;
    w1T[c * D_ + r] = w1[r * D_ + c];
    w2T[c * D_ + r] = w2[r * D_ + c];
}

// Kernel 1: wf = L2normalize(feat * sigmoid(weights)) along D.
// One wave per node (8 nodes / 256-thread block), lane handles d and d+32.
__global__ __launch_bounds__(256) void wf_kernel(const float* __restrict__ feat,
                                                 const float* __restrict__ w,
                                                 float* __restrict__ wf) {
    const int wave = threadIdx.x >> 5, lane = threadIdx.x & 31;
    const int node = blockIdx.x * 8 + wave;           // grid sized exactly: B*T*N/8
    const float s0 = 1.f / (1.f + expf(-w[lane]));
    const float s1 = 1.f / (1.f + expf(-w[lane + 32]));
    const float* src = feat + (size_t)node * D_;
    float x0 = src[lane] * s0;
    float x1 = src[lane + 32] * s1;
    float ss = x0 * x0 + x1 * x1;
    #pragma unroll
    for (int m = 16; m >= 1; m >>= 1) ss += __shfl_xor(ss, m, 32);
    const float sc = 1.f / fmaxf(sqrtf(ss), 1e-12f);
    float* dst = wf + (size_t)node * D_;
    dst[lane]      = x0 * sc;
    dst[lane + 32] = x1 * sc;
}

// Kernel 2: per (b,t2) workgroup.
// Phase 1: M[64x64] = Wwin^T * Win over 900 window rows (WMMA f32 16x16x4), wsum[64].
// Phase 2: per 16-node tile: X=A*M, deg=A*wsum, agg=X/deg, FFN (w1/relu/w2), residual+LN.
__global__ __launch_bounds__(256) void gnn_main(const float* __restrict__ feat,
                                                const float* __restrict__ wf,
                                                const float* __restrict__ w1T,
                                                const float* __restrict__ b1,
                                                const float* __restrict__ w2T,
                                                const float* __restrict__ b2,
                                                const float* __restrict__ gamma,
                                                const float* __restrict__ beta,
                                                float* __restrict__ out) {
    __shared__ float sm[13376];               // 53.5 KB
    float* const R    = sm;                   // 13056-float region, phase-aliased:
    float* const CHWT = R;                    //  phase1: wwin chunk, TRANSPOSED [64 feat][PAD]
    float* const CHXT = R + 64 * PAD;         //  phase1: win  chunk, TRANSPOSED [64 feat][PAD]
    float* const MST  = R;                    //  phase2: M TRANSPOSED [64 j][PAD] (MST[j][k])
    float* const SCR  = R + 64 * PAD;         //  phase2: per-wave scratch 8 x [16][PAD]
    float* const wsumS = sm + 13056;
    float* const b1S = wsumS + 64;
    float* const b2S = b1S + 64;
    float* const gS  = b2S + 64;
    float* const btS = gS + 64;

    const int tid = threadIdx.x;
    const int wave = tid >> 5, lane = tid & 31;
    const int h = lane >> 4, ml = lane & 15;
    const int b = blockIdx.x / T2_, t2 = blockIdx.x % T2_;

    if (tid < D_) { b1S[tid] = b1[tid]; b2S[tid] = b2[tid]; gS[tid] = gamma[tid]; btS[tid] = beta[tid]; }

    // ---------------- Phase 1: M = Wwin^T * Win, wsum = colsum(Wwin) ----------------
    const int mi = wave & 3;            // M-row tile (wwin feature dim)
    const int jb = (wave >> 2) * 2;     // two N-col tiles (win feature dim)
    v8f c0 = {}, c1 = {};
    float wsl = 0.f;

    for (int t = 0; t < 3; ++t) {
        for (int cck = 0; cck < 5; ++cck) {
            __syncthreads();            // previous chunk fully consumed
            const int base = ((b * T_ + t2 + t) * N_ + cck * 60) * D_;
            if (tid == 0 && (t * 5 + cck) < 14) {   // warm next chunk
                const int nb = base + 60 * D_;
                __builtin_prefetch(wf + nb, 0, 0);
                __builtin_prefetch(feat + nb, 0, 0);
            }
            // row-major coalesced float4 global reads, transposed scatter into LDS
            for (int i = tid; i < 60 * 16; i += 256) {
                const int r = i >> 4, q = i & 15;
                const float4 a = *(const float4*)(wf   + base + r * D_ + q * 4);
                const float4 x = *(const float4*)(feat + base + r * D_ + q * 4);
                CHWT[(q * 4 + 0) * PAD + r] = a.x;
                CHWT[(q * 4 + 1) * PAD + r] = a.y;
                CHWT[(q * 4 + 2) * PAD + r] = a.z;
                CHWT[(q * 4 + 3) * PAD + r] = a.w;
                CHXT[(q * 4 + 0) * PAD + r] = x.x;
                CHXT[(q * 4 + 1) * PAD + r] = x.y;
                CHXT[(q * 4 + 2) * PAD + r] = x.z;
                CHXT[(q * 4 + 3) * PAD + r] = x.w;
            }
            __syncthreads();
            const float* rowA  = CHWT + (mi * 16 + ml) * PAD;
            const float* rowB0 = CHXT + (jb * 16 + ml) * PAD;
            const float* rowB1 = rowB0 + 16 * PAD;
            for (int ks = 0; ks < 15; ++ks) {       // K = 60 rows, 4 per WMMA
                const int kr = ks * 4 + 2 * h;
                v2f a  = *(const v2f*)(rowA  + kr);   // single b64, no packing
                v2f v0 = *(const v2f*)(rowB0 + kr);
                v2f v1 = *(const v2f*)(rowB1 + kr);
                c0 = wmma4(a, v0, c0);
                c1 = wmma4(a, v1, c1);
            }
            if (tid < D_) {                         // waves 0-1: accumulate wsum (float4 reads)
                const float* rw = CHWT + tid * PAD;
                for (int r = 0; r < 60; r += 4) {
                    const float4 v = *(const float4*)(rw + r);
                    wsl += v.x + v.y + v.z + v.w;
                }
            }
        }
    }
    __syncthreads();                                // done reading chunks; region becomes MST/SCR
    #pragma unroll
    for (int vr = 0; vr < 8; ++vr) {                // C/D layout -> LDS M^T[j][k]
        const int row = mi * 16 + vr + 8 * h;       // k index
        MST[(jb * 16 + ml) * PAD + row]       = c0[vr];
        MST[((jb + 1) * 16 + ml) * PAD + row] = c1[vr];
    }
    if (tid < D_) wsumS[tid] = wsl;
    __syncthreads();

    // ---------------- Phase 2: per 16-node tile chain ----------------
    const int fbase = (b * T_ + t2 + 2) * N_;       // current-timestep nodes
    const int obase = (b * T2_ + t2) * N_;
    float* const SW = SCR + wave * (16 * PAD);

    for (int tile = wave; tile < NTILE; tile += 8) {
        const int n0 = tile * 16;
        // stage wf node-tile into per-wave scratch (row-major; A-operand pairs contiguous)
        for (int i = lane; i < 256; i += 32) {
            const int r = i >> 4, q = i & 15;
            int nd = n0 + r; if (nd > N_ - 1) nd = N_ - 1;
            *(float4*)(SW + r * PAD + q * 4) =
                *(const float4*)(wf + (size_t)(fbase + nd) * D_ + q * 4);
        }
        v2f A[16];
        #pragma unroll
        for (int kk = 0; kk < 16; ++kk) A[kk] = *(const v2f*)(SW + ml * PAD + 4 * kk + 2 * h);

        // deg = a . wsum  (row stays zero if deg == 0, as in reference)
        float dg = 0.f;
        #pragma unroll
        for (int kk = 0; kk < 16; ++kk) {
            const int ck = 4 * kk + 2 * h;
            dg += A[kk].x * wsumS[ck] + A[kk].y * wsumS[ck + 1];
        }
        dg += __shfl_xor(dg, 16, 32);
        const float dinv = (dg != 0.f) ? (1.f / dg) : 0.f;

        // X = A * M   (B-operand = contiguous v2f from M^T)
        v8f CC[4];
        #pragma unroll
        for (int ji = 0; ji < 4; ++ji) {
            v8f acc = {};
            const float* mrow = MST + (ji * 16 + ml) * PAD;
            #pragma unroll
            for (int kk = 0; kk < 16; ++kk) {
                v2f bv = *(const v2f*)(mrow + 4 * kk + 2 * h);
                acc = wmma4(A[kk], bv, acc);
            }
            CC[ji] = acc;
        }
        // agg = X * dinv  -> scratch (C-layout store), reload in A-layout
        #pragma unroll
        for (int vr = 0; vr < 8; ++vr) {
            const float dv = __shfl(dinv, vr + 8 * h, 32);
            const int row = (vr + 8 * h) * PAD;
            #pragma unroll
            for (int ji = 0; ji < 4; ++ji) SW[row + ji * 16 + ml] = CC[ji][vr] * dv;
        }
        #pragma unroll
        for (int kk = 0; kk < 16; ++kk) A[kk] = *(const v2f*)(SW + ml * PAD + 4 * kk + 2 * h);

        // H1 = relu(agg * w1 + b1)   (B-operand = contiguous v2f from w1^T in global)
        #pragma unroll
        for (int ji = 0; ji < 4; ++ji) {
            v8f acc = {};
            const float* wrow = w1T + (ji * 16 + ml) * D_;
            #pragma unroll
            for (int kk = 0; kk < 16; ++kk) {
                v2f bv = *(const v2f*)(wrow + 4 * kk + 2 * h);
                acc = wmma4(A[kk], bv, acc);
            }
            const float bb = b1S[ji * 16 + ml];
            #pragma unroll
            for (int vr = 0; vr < 8; ++vr) acc[vr] = fmaxf(acc[vr] + bb, 0.f);
            CC[ji] = acc;
        }
        #pragma unroll
        for (int vr = 0; vr < 8; ++vr) {
            const int row = (vr + 8 * h) * PAD;
            #pragma unroll
            for (int ji = 0; ji < 4; ++ji) SW[row + ji * 16 + ml] = CC[ji][vr];
        }
        #pragma unroll
        for (int kk = 0; kk < 16; ++kk) A[kk] = *(const v2f*)(SW + ml * PAD + 4 * kk + 2 * h);

        // H2 = H1 * w2 + b2
        #pragma unroll
        for (int ji = 0; ji < 4; ++ji) {
            v8f acc = {};
            const float* wrow = w2T + (ji * 16 + ml) * D_;
            #pragma unroll
            for (int kk = 0; kk < 16; ++kk) {
                v2f bv = *(const v2f*)(wrow + 4 * kk + 2 * h);
                acc = wmma4(A[kk], bv, acc);
            }
            const float bb = b2S[ji * 16 + ml];
            #pragma unroll
            for (int vr = 0; vr < 8; ++vr) acc[vr] += bb;
            CC[ji] = acc;
        }

        // residual + LayerNorm + store (row stats via wave32 xor-shuffles within 16-lane halves)
        #pragma unroll
        for (int vr = 0; vr < 8; ++vr) {
            const int node = n0 + vr + 8 * h;
            const int ndc = node > N_ - 1 ? N_ - 1 : node;
            const float* fr = feat + (size_t)(fbase + ndc) * D_;
            float s0 = CC[0][vr] + fr[ml];
            float s1 = CC[1][vr] + fr[16 + ml];
            float s2 = CC[2][vr] + fr[32 + ml];
            float s3 = CC[3][vr] + fr[48 + ml];
            float p  = s0 + s1 + s2 + s3;
            float q2 = s0 * s0 + s1 * s1 + s2 * s2 + s3 * s3;
            #pragma unroll
            for (int mk = 1; mk <= 8; mk <<= 1) {
                p  += __shfl_xor(p,  mk, 32);
                q2 += __shfl_xor(q2, mk, 32);
            }
            const float mean = p * 0.015625f;
            const float var  = q2 * 0.015625f - mean * mean;
            const float rstd = rsqrtf(var + 1e-5f);
            if (node < N_) {
                float* o = out + (size_t)(obase + node) * D_;
                o[ml]      = (s0 - mean) * rstd * gS[ml]      + btS[ml];
                o[16 + ml] = (s1 - mean) * rstd * gS[16 + ml] + btS[16 + ml];
                o[32 + ml] = (s2 - mean) * rstd * gS[32 + ml] + btS[32 + ml];
                o[48 + ml] = (s3 - mean) * rstd * gS[48 + ml] + btS[48 + ml];
            }
        }
    }
}

extern "C" void kernel_launch(void* const* d_in, const int* in_sizes, int n_in,
                              void* d_out, int out_size, void* d_ws, size_t ws_size,
                              hipStream_t stream) {
    const float* feat    = (const float*)d_in[0];
    const float* weights = (const float*)d_in[1];
    const float* w1      = (const float*)d_in[2];
    const float* b1      = (const float*)d_in[3];
    const float* w2      = (const float*)d_in[4];
    const float* b2      = (const float*)d_in[5];
    const float* gamma   = (const float*)d_in[6];
    const float* beta    = (const float*)d_in[7];
    float* out = (float*)d_out;

    float* wf  = (float*)d_ws;                       // B*T*N*D floats = 14.06 MB
    float* w1T = wf + (size_t)B_ * T_ * N_ * D_;     // 4096 floats
    float* w2T = w1T + D_ * D_;                      // 4096 floats

    // Kernel 0: transpose FFN weights (contiguous WMMA B-operands)
    wT_kernel<<<16, 256, 0, stream>>>(w1, w2, w1T, w2T);
    // Kernel 1: gated + L2-normalized features (all timesteps)
    wf_kernel<<<(B_ * T_ * N_) / 8, 256, 0, stream>>>(feat, weights, wf);
    // Kernel 2: one workgroup per (b, t2)
    gnn_main<<<B_ * T2_, 256, 0, stream>>>(feat, wf, w1T, b1, w2T, b2, gamma, beta, out);
}